// FeedbackLoss_5368709120151
// MI455X (gfx1250) — compile-verified
//
#include <hip/hip_runtime.h>
#include <hip/hip_bf16.h>

#define NSEG 1024
#define NBATCH 16
#define MIN_PIX 8.0f
#define EPSF 1e-6f

typedef __attribute__((ext_vector_type(2))) float v2f;
typedef __attribute__((ext_vector_type(8))) float v8f;

// ---------------------------------------------------------------------------
// Phase 0: zero the workspace accumulators (harness poisons d_ws with 0xAA).
// ---------------------------------------------------------------------------
__global__ void fb_zero_ws(float* __restrict__ ws, int n) {
    int i = blockIdx.x * blockDim.x + threadIdx.x;
    if (i < n) ws[i] = 0.0f;
}

// ---------------------------------------------------------------------------
// Phase 1: streaming segmented sums.  Each WG owns a contiguous chunk of one
// batch image, accumulates S=1024 bins in LDS via ds_add_f32, then flushes
// non-empty bins to the global per-(b,s) table with global atomics.
// Loads are float4/int4 (GLOBAL_LOAD_B128) to maximize HBM efficiency;
// global_prefetch hints issued one block-stride ahead.
// ---------------------------------------------------------------------------
__global__ __launch_bounds__(256)
void fb_seg_sums(const float4* __restrict__ pred4,
                 const float4* __restrict__ clean4,
                 const int4*  __restrict__ inst4,
                 float* __restrict__ wsPred,
                 float* __restrict__ wsClean,
                 float* __restrict__ wsCount,
                 int vecPerBatch, int wgsPerBatch) {
    __shared__ float sPred[NSEG];
    __shared__ float sClean[NSEG];
    __shared__ float sCnt[NSEG];

    const int tid = threadIdx.x;
    for (int i = tid; i < NSEG; i += 256) {
        sPred[i] = 0.0f; sClean[i] = 0.0f; sCnt[i] = 0.0f;
    }
    __syncthreads();

    const int b     = blockIdx.x / wgsPerBatch;
    const int wg    = blockIdx.x - b * wgsPerBatch;
    const int chunk = vecPerBatch / wgsPerBatch;
    const long start = (long)b * vecPerBatch + (long)wg * chunk;

    for (int i = tid; i < chunk; i += 256) {
        const long idx = start + i;
        // speculative prefetch one block-stride ahead -> global_prefetch_b8
        __builtin_prefetch(&pred4[idx + 256], 0, 0);
        __builtin_prefetch(&clean4[idx + 256], 0, 0);
        __builtin_prefetch(&inst4[idx + 256], 0, 0);

        const float4 p = pred4[idx];
        const float4 c = clean4[idx];
        const int4   s = inst4[idx];

        atomicAdd(&sPred[s.x], p.x); atomicAdd(&sClean[s.x], c.x); atomicAdd(&sCnt[s.x], 1.0f);
        atomicAdd(&sPred[s.y], p.y); atomicAdd(&sClean[s.y], c.y); atomicAdd(&sCnt[s.y], 1.0f);
        atomicAdd(&sPred[s.z], p.z); atomicAdd(&sClean[s.z], c.z); atomicAdd(&sCnt[s.z], 1.0f);
        atomicAdd(&sPred[s.w], p.w); atomicAdd(&sClean[s.w], c.w); atomicAdd(&sCnt[s.w], 1.0f);
    }
    __syncthreads();

    const int base = b * NSEG;
    for (int i = tid; i < NSEG; i += 256) {
        const float n = sCnt[i];
        if (n != 0.0f) {
            atomicAdd(&wsCount[base + i], n);
            atomicAdd(&wsPred[base + i],  sPred[i]);
            atomicAdd(&wsClean[base + i], sClean[i]);
        }
    }
}

// ---------------------------------------------------------------------------
// Phase 2: one wave32.  Per-cell BCE / masks, then per-batch sums over the
// 1024-segment axis via chained V_WMMA_F32_16X16X4_F32 with B = ones:
//   D[b, *] += sum_k A[b, k]      (exact f32, batch b == matrix row M)
// Six independent accumulators: valid, bce*valid, tumor_n, tumor_conf,
// normal_n, normal_conf.  Per ISA layout, after the loop lane0 VGPR r holds
// batch r (N=0) and lane16 VGPR r holds batch r+8.
// ---------------------------------------------------------------------------
__device__ __forceinline__ void fb_cell_q(float cnt, float ps, float cs, int s,
                                          float q[6]) {
    const float denom = fmaxf(cnt, 1.0f);
    const float pm = ps / denom;               // pred_mean
    const float t  = cs / denom;               // target_mean
    const float vf = (cnt >= MIN_PIX && s > 0) ? 1.0f : 0.0f;
    const float p  = fminf(fmaxf(pm, EPSF), 1.0f - EPSF);
    const float bce = -(t * logf(p) + (1.0f - t) * log1pf(-p));
    const float tum = vf * ((t > 0.5f) ? 1.0f : 0.0f);
    const float nrm = vf * ((t <= 0.5f) ? 1.0f : 0.0f);
    q[0] = vf;
    q[1] = bce * vf;
    q[2] = tum;
    q[3] = pm * tum;
    q[4] = nrm;
    q[5] = pm * nrm;
}

__global__ __launch_bounds__(32)
void fb_finalize(const float* __restrict__ wsPred,
                 const float* __restrict__ wsClean,
                 const float* __restrict__ wsCount,
                 float* __restrict__ out) {
    const int lane = threadIdx.x;      // 0..31, EXEC all ones for WMMA
    const int mrow = lane & 15;        // batch index == A-matrix row M
    const int half = lane >> 4;        // 0 -> K{0,1}, 1 -> K{2,3}

    v8f acc0 = {}; v8f acc1 = {}; v8f acc2 = {};
    v8f acc3 = {}; v8f acc4 = {}; v8f acc5 = {};
    v2f bones; bones.x = 1.0f; bones.y = 1.0f;   // B = ones (layout-invariant)

    for (int s0 = 0; s0 < NSEG; s0 += 4) {
        const int sA = s0 + 2 * half;      // this lane's K=even segment
        const int sB = sA + 1;             // K=odd segment
        const int iA = mrow * NSEG + sA;
        const int iB = iA + 1;

        float qa[6], qb[6];
        fb_cell_q(wsCount[iA], wsPred[iA], wsClean[iA], sA, qa);
        fb_cell_q(wsCount[iB], wsPred[iB], wsClean[iB], sB, qb);

        v2f a;
        a.x = qa[0]; a.y = qb[0];
        acc0 = __builtin_amdgcn_wmma_f32_16x16x4_f32(false, a, false, bones,
                                                     (short)0, acc0, false, false);
        a.x = qa[1]; a.y = qb[1];
        acc1 = __builtin_amdgcn_wmma_f32_16x16x4_f32(false, a, false, bones,
                                                     (short)0, acc1, false, false);
        a.x = qa[2]; a.y = qb[2];
        acc2 = __builtin_amdgcn_wmma_f32_16x16x4_f32(false, a, false, bones,
                                                     (short)0, acc2, false, false);
        a.x = qa[3]; a.y = qb[3];
        acc3 = __builtin_amdgcn_wmma_f32_16x16x4_f32(false, a, false, bones,
                                                     (short)0, acc3, false, false);
        a.x = qa[4]; a.y = qb[4];
        acc4 = __builtin_amdgcn_wmma_f32_16x16x4_f32(false, a, false, bones,
                                                     (short)0, acc4, false, false);
        a.x = qa[5]; a.y = qb[5];
        acc5 = __builtin_amdgcn_wmma_f32_16x16x4_f32(false, a, false, bones,
                                                     (short)0, acc5, false, false);
    }

    // Extract column N=0: lane0 -> batches 0..7, lane16 -> batches 8..15.
    __shared__ float red[NBATCH][6];
    if (lane == 0) {
#pragma unroll
        for (int r = 0; r < 8; ++r) {
            red[r][0] = acc0[r]; red[r][1] = acc1[r]; red[r][2] = acc2[r];
            red[r][3] = acc3[r]; red[r][4] = acc4[r]; red[r][5] = acc5[r];
        }
    }
    if (lane == 16) {
#pragma unroll
        for (int r = 0; r < 8; ++r) {
            red[r + 8][0] = acc0[r]; red[r + 8][1] = acc1[r]; red[r + 8][2] = acc2[r];
            red[r + 8][3] = acc3[r]; red[r + 8][4] = acc4[r]; red[r + 8][5] = acc5[r];
        }
    }
    __syncthreads();

    if (lane == 0) {
        float vc = 0.0f, lsum = 0.0f;
        float nbT = 0.0f, cT = 0.0f;
        float nbN = 0.0f, cN = 0.0f;
        for (int b = 0; b < NBATCH; ++b) {
            const float nv = red[b][0];
            const float lb = red[b][1] / fmaxf(nv, 1.0f);
            const float bv = (nv > 0.0f) ? 1.0f : 0.0f;
            vc += bv; lsum += lb * bv;

            const float nt = red[b][2];
            const float ct = red[b][3] / fmaxf(nt, 1.0f);
            const float ht = (nt > 0.0f) ? 1.0f : 0.0f;
            nbT += ht; cT += ct * ht;

            const float nn = red[b][4];
            const float cn = red[b][5] / fmaxf(nn, 1.0f);
            const float hn = (nn > 0.0f) ? 1.0f : 0.0f;
            nbN += hn; cN += cn * hn;
        }
        out[0] = (vc > 0.0f) ? lsum / fmaxf(vc, 1.0f) : 0.0f;
        out[1] = 0.0f;  // loss_entropy
        out[2] = (vc > 0.0f && nbT > 0.0f) ? cT / fmaxf(nbT, 1.0f) : -1.0f;
        out[3] = (vc > 0.0f && nbN > 0.0f) ? cN / fmaxf(nbN, 1.0f) : -1.0f;
    }
}

// ---------------------------------------------------------------------------
extern "C" void kernel_launch(void* const* d_in, const int* in_sizes, int n_in,
                              void* d_out, int out_size, void* d_ws, size_t ws_size,
                              hipStream_t stream) {
    const float* pred  = (const float*)d_in[0];
    const float* clean = (const float*)d_in[1];
    const int*   inst  = (const int*)d_in[2];
    float* out = (float*)d_out;
    float* ws  = (float*)d_ws;

    const int totalPix    = in_sizes[0];           // B*H*W = 16M
    const int pixPerBatch = totalPix / NBATCH;     // 1M
    const int vecPerBatch = pixPerBatch / 4;       // float4 vectors
    const int WGS_PER_BATCH = 32;                  // 512 WGs total

    float* wsPred  = ws;
    float* wsClean = ws + NBATCH * NSEG;
    float* wsCount = ws + 2 * NBATCH * NSEG;

    const int nz = 3 * NBATCH * NSEG;
    fb_zero_ws<<<(nz + 255) / 256, 256, 0, stream>>>(ws, nz);

    fb_seg_sums<<<NBATCH * WGS_PER_BATCH, 256, 0, stream>>>(
        (const float4*)pred, (const float4*)clean, (const int4*)inst,
        wsPred, wsClean, wsCount, vecPerBatch, WGS_PER_BATCH);

    fb_finalize<<<1, 32, 0, stream>>>(wsPred, wsClean, wsCount, out);
}